// EdgeClassify_79929341378653
// MI455X (gfx1250) — compile-verified
//
#include <hip/hip_runtime.h>

#define BATCH  8
#define MSP    2048
#define DMODEL 512
#define AMAX   64
#define NBONDS 5

#define KS     32    // K chunk staged in LDS
#define DTILE  32    // D columns per block
#define WL_STR 36    // LDS row stride (floats) for W tile: mult-of-4 (16B stores), conflict-free A reads
#define E_STR  48    // LDS row stride (floats) for e tile: halves land in disjoint bank sets

typedef __attribute__((ext_vector_type(2))) float v2f;
typedef __attribute__((ext_vector_type(4))) float v4f;
typedef __attribute__((ext_vector_type(8))) float v8f;

// Kernel 1: reduce_e[b,a,d] = sum_s W_ll[a,s] * e[b,s,d] + b_ll[a]
// One block per (batch, 32-wide D strip); 8 waves, each owns one 16x16 C tile.
__global__ __launch_bounds__(256) void
reduce_gemm_kernel(const float* __restrict__ e, const float* __restrict__ Wll,
                   const float* __restrict__ bll, float* __restrict__ red) {
  __shared__ float sW[AMAX * WL_STR];   // 64 x 36 floats  (9216 B)
  __shared__ float sE[KS * E_STR];      // 32 x 48 floats  (6144 B)

  const int tid  = threadIdx.x;
  const int lane = tid & 31;
  const int l15  = lane & 15;
  const int half = lane >> 4;           // K-half select for wave32 fragments
  const int wv   = tid >> 5;            // wave id 0..7
  const int rt   = wv & 3;              // A row-tile (16 rows each)
  const int ct   = wv >> 2;             // 16-col tile within the 32-wide D strip

  const int bx   = blockIdx.x;
  const int bIdx = bx >> 4;             // / (DMODEL/DTILE) == /16
  const int d0   = (bx & 15) * DTILE;

  // cooperative staging indices (float4 granularity)
  const int wRow0 = tid >> 3;           // rows 0..31 of W chunk
  const int wRow1 = (tid + 256) >> 3;   // rows 32..63 of W chunk
  const int wC4   = (tid & 7) * 4;
  const int eRow  = tid >> 3;           // rows 0..31 of e chunk
  const int eC4   = (tid & 7) * 4;

  const float* eB = e + (size_t)bIdx * MSP * DMODEL;

  v8f acc = {0.f, 0.f, 0.f, 0.f, 0.f, 0.f, 0.f, 0.f};

  for (int k0 = 0; k0 < MSP; k0 += KS) {
    // stage W_ll[0:64, k0:k0+32] and e[b, k0:k0+32, d0:d0+32] into LDS (coalesced b128)
    v4f w0 = *(const v4f*)(Wll + (size_t)wRow0 * MSP + k0 + wC4);
    v4f w1 = *(const v4f*)(Wll + (size_t)wRow1 * MSP + k0 + wC4);
    v4f ev = *(const v4f*)(eB + (size_t)(k0 + eRow) * DMODEL + d0 + eC4);
    *(v4f*)(sW + wRow0 * WL_STR + wC4) = w0;
    *(v4f*)(sW + wRow1 * WL_STR + wC4) = w1;
    *(v4f*)(sE + eRow * E_STR + eC4)   = ev;

    if (k0 + KS < MSP) {  // uniform branch: pull next chunk toward the caches
      __builtin_prefetch(Wll + (size_t)wRow0 * MSP + k0 + KS + wC4, 0, 3);
      __builtin_prefetch(eB + (size_t)(k0 + KS + eRow) * DMODEL + d0 + eC4, 0, 3);
    }
    __syncthreads();

#pragma unroll
    for (int kk = 0; kk < KS; kk += 4) {
      // A fragment (16x4 f32): lane l -> row rt*16+l15, K = kk + half*2 + {0,1}
      v2f a = *(const v2f*)(sW + (rt * 16 + l15) * WL_STR + kk + half * 2);
      // B fragment (4x16 f32): v0 = K row (kk+half*2), v1 = K row (kk+half*2+1), N = ct*16+l15
      v2f b;
      b.x = sE[(kk + half * 2 + 0) * E_STR + ct * 16 + l15];
      b.y = sE[(kk + half * 2 + 1) * E_STR + ct * 16 + l15];
      acc = __builtin_amdgcn_wmma_f32_16x16x4_f32(false, a, false, b,
                                                  (short)0, acc, false, false);
    }
    __syncthreads();
  }

  // C/D layout: VGPR r holds M = r (lanes 0-15) / r+8 (lanes 16-31), N = l15
#pragma unroll
  for (int r = 0; r < 8; ++r) {
    int arow = rt * 16 + r + half * 8;
    int dcol = d0 + ct * 16 + l15;
    red[((size_t)bIdx * AMAX + arow) * DMODEL + dcol] = acc[r] + bll[arow];
  }
}

// Kernel 2: out[b,a,nb] = sum_d red[b,a,d]*(W_fl[nb,d]+W_fl[nb,D+d]) + b_fl[nb],
// broadcast along j. One wave32 per (b,a).
__global__ __launch_bounds__(256) void
pair_broadcast_kernel(const float* __restrict__ red, const float* __restrict__ Wfl,
                      const float* __restrict__ bfl, float* __restrict__ out) {
  const int tid  = threadIdx.x;
  const int lane = tid & 31;
  const int gw   = blockIdx.x * 8 + (tid >> 5);  // 0..511
  const int b    = gw >> 6;
  const int a    = gw & (AMAX - 1);

  const float* row = red + ((size_t)b * AMAX + a) * DMODEL;

  float s0 = 0.f, s1 = 0.f, s2 = 0.f, s3 = 0.f, s4 = 0.f;
  for (int d = lane; d < DMODEL; d += 32) {
    float x = row[d];
    float w0 = Wfl[0 * 2 * DMODEL + d] + Wfl[0 * 2 * DMODEL + DMODEL + d];
    float w1 = Wfl[1 * 2 * DMODEL + d] + Wfl[1 * 2 * DMODEL + DMODEL + d];
    float w2 = Wfl[2 * 2 * DMODEL + d] + Wfl[2 * 2 * DMODEL + DMODEL + d];
    float w3 = Wfl[3 * 2 * DMODEL + d] + Wfl[3 * 2 * DMODEL + DMODEL + d];
    float w4 = Wfl[4 * 2 * DMODEL + d] + Wfl[4 * 2 * DMODEL + DMODEL + d];
    s0 = fmaf(x, w0, s0); s1 = fmaf(x, w1, s1); s2 = fmaf(x, w2, s2);
    s3 = fmaf(x, w3, s3); s4 = fmaf(x, w4, s4);
  }
#pragma unroll
  for (int m = 16; m >= 1; m >>= 1) {   // wave32 butterfly: every lane ends with the sum
    s0 += __shfl_xor(s0, m, 32);
    s1 += __shfl_xor(s1, m, 32);
    s2 += __shfl_xor(s2, m, 32);
    s3 += __shfl_xor(s3, m, 32);
    s4 += __shfl_xor(s4, m, 32);
  }
  s0 += bfl[0]; s1 += bfl[1]; s2 += bfl[2]; s3 += bfl[3]; s4 += bfl[4];

  float* dst = out + (size_t)(b * AMAX + a) * AMAX * NBONDS;  // 320 contiguous floats
  for (int t = lane; t < AMAX * NBONDS; t += 32) {
    int m5 = t % NBONDS;
    float v = s0;
    v = (m5 == 1) ? s1 : v;
    v = (m5 == 2) ? s2 : v;
    v = (m5 == 3) ? s3 : v;
    v = (m5 == 4) ? s4 : v;
    dst[t] = v;
  }
}

extern "C" void kernel_launch(void* const* d_in, const int* in_sizes, int n_in,
                              void* d_out, int out_size, void* d_ws, size_t ws_size,
                              hipStream_t stream) {
  const float* e   = (const float*)d_in[0];  // [B, MSP, D]
  const float* Wll = (const float*)d_in[1];  // [A, MSP]
  const float* bll = (const float*)d_in[2];  // [A]
  const float* Wfl = (const float*)d_in[3];  // [NB, 2D]
  const float* bfl = (const float*)d_in[4];  // [NB]
  float* red = (float*)d_ws;                 // reduce_e [B, A, D] fp32 = 1 MB scratch
  float* out = (float*)d_out;                // [B, A, A, NB]

  reduce_gemm_kernel<<<dim3(BATCH * (DMODEL / DTILE)), dim3(256), 0, stream>>>(
      e, Wll, bll, red);
  pair_broadcast_kernel<<<dim3(BATCH * AMAX / 8), dim3(256), 0, stream>>>(
      red, Wfl, bfl, out);
}